// RelativePositionalEmbedding2D_84688165143031
// MI455X (gfx1250) — compile-verified
//
#include <hip/hip_runtime.h>

typedef __attribute__((ext_vector_type(2))) float v2f;
typedef __attribute__((ext_vector_type(4))) float v4f;
typedef __attribute__((ext_vector_type(8))) float v8f;

#define NB      2
#define NHEADS  8
#define TQ      8
#define HQ      16
#define WQ      16
#define CDIM    64
#define QTOK    (TQ * HQ * WQ)          /* 2048 */
#define KTOK    2048
#define NROWS   (NB * NHEADS * QTOK)    /* 32768 */

// ---------------------------------------------------------------------------
// Phase 1: materialize rel_h / rel_w / rel_t with V_WMMA_F32_16X16X4_F32.
// One wave = one 16x16 output tile, K=64 accumulated over 16 chained WMMAs.
//   mode 0 (tiles 0..2047):    rows = w-contiguous (fixed b,N,t,h) -> rel_h, rel_t
//   mode 1 (tiles 2048..4095): rows = h-strided    (fixed b,N,t,w) -> rel_w
// A layout (16x4 f32): lane&15 = M, lane>>4 selects K-half; 2 floats/lane/step.
// B layout (4x16 f32): lane&15 = N, lane>>4 selects K-half; 2 floats/lane/step.
// C/D layout: col N = lane&15, row M = vgpr + 8*(lane>>4).
// ---------------------------------------------------------------------------
__global__ __launch_bounds__(256)
void rel_wmma_kernel(const float* __restrict__ q,
                     const float* __restrict__ hemb,
                     const float* __restrict__ wemb,
                     const float* __restrict__ temb,
                     float* __restrict__ relh,
                     float* __restrict__ relw,
                     float* __restrict__ relt)
{
    const int lane  = threadIdx.x & 31;
    const int wave  = threadIdx.x >> 5;
    const int tile  = blockIdx.x * 8 + wave;   // 0..4095
    const int mode  = tile >> 11;              // 0 or 1
    const int tid   = tile & 2047;
    const int i16   = tid & 15;                // h (mode 0) / w (mode 1)
    const int t     = (tid >> 4) & 7;
    const int bn    = tid >> 7;                // b*8 + head

    const int m     = lane & 15;               // A row M  (and B col N)
    const int khalf = lane >> 4;               // K-half select

    const int  rowStride = (mode == 0) ? CDIM : (WQ * CDIM);
    const long qbase = ((long)bn * QTOK + t * (HQ * WQ) +
                        (mode == 0 ? i16 * WQ : i16)) * CDIM;
    const float* arow = q + qbase + (long)m * rowStride;

    // primary B table: Rh (mode 0) or Rw (mode 1); index = i16 - nk + 15
    const float* prim = (mode == 0) ? hemb : wemb;
    const float* brow = prim + (i16 - m + 15) * CDIM;
    // time table (mode 0 only); pad key columns 8..15 with zeros
    const int   tclamp = (m < 8) ? m : 7;
    const float tmask  = (m < 8) ? 1.0f : 0.0f;
    const float* trow  = temb + (t - tclamp + 7) * CDIM;

    v8f c0 = {};
    v8f c1 = {};

    if (mode == 0) {
        #pragma unroll
        for (int kk = 0; kk < 16; ++kk) {
            const int ko = kk * 4 + khalf * 2;
            v2f a  = *(const v2f*)(arow + ko);
            v2f bh = *(const v2f*)(brow + ko);
            v2f bt = *(const v2f*)(trow + ko);
            bt.x *= tmask; bt.y *= tmask;
            c0 = __builtin_amdgcn_wmma_f32_16x16x4_f32(
                     false, a, false, bh, (short)0, c0, false, false);
            c1 = __builtin_amdgcn_wmma_f32_16x16x4_f32(
                     false, a, false, bt, (short)0, c1, false, false);
        }
        const int col = m;
        #pragma unroll
        for (int i = 0; i < 8; ++i) {
            const int M = i + 8 * khalf;                       // query row in tile (= w)
            const long r = (long)bn * QTOK + t * 256 + i16 * 16 + M;
            relh[r * 16 + col] = c0[i];
            if (col < 8) relt[r * 8 + col] = c1[i];
        }
    } else {
        #pragma unroll
        for (int kk = 0; kk < 16; ++kk) {
            const int ko = kk * 4 + khalf * 2;
            v2f a  = *(const v2f*)(arow + ko);
            v2f bw = *(const v2f*)(brow + ko);
            c0 = __builtin_amdgcn_wmma_f32_16x16x4_f32(
                     false, a, false, bw, (short)0, c0, false, false);
        }
        const int col = m;
        #pragma unroll
        for (int i = 0; i < 8; ++i) {
            const int M = i + 8 * khalf;                       // query row in tile (= h)
            const long r = (long)bn * QTOK + t * 256 + M * 16 + i16;
            relw[r * 16 + col] = c0[i];
        }
    }
}

// ---------------------------------------------------------------------------
// Phase 2: out = scores + relh[hk] + relw[wk] + relt[tk]   (pure streaming)
// One block per query row: 2048 key elements = 512 threads x float4.
// 40 bias floats staged in LDS; non-temporal 128-bit stream for the 512 MB.
// ---------------------------------------------------------------------------
__global__ __launch_bounds__(512)
void add_bias_kernel(const float* __restrict__ scores,
                     const float* __restrict__ relh,
                     const float* __restrict__ relw,
                     const float* __restrict__ relt,
                     float* __restrict__ out)
{
    __shared__ __align__(16) float sh[16];
    __shared__ __align__(16) float sw[16];
    __shared__ __align__(16) float st[8];

    const int row = blockIdx.x;            // 0..32767
    const int tid = threadIdx.x;           // 0..511

    if (tid < 16)       sh[tid]      = relh[(long)row * 16 + tid];
    else if (tid < 32)  sw[tid - 16] = relw[(long)row * 16 + (tid - 16)];
    else if (tid < 40)  st[tid - 32] = relt[(long)row * 8  + (tid - 32)];
    __syncthreads();

    const int j   = tid * 4;               // key index (multiple of 4)
    const int tk  = j >> 8;
    const int hk  = (j >> 4) & 15;
    const int wkb = j & 15;                // 4-aligned within [0,16)

    const long off = (long)row * KTOK + j;
    const v4f s  = __builtin_nontemporal_load((const v4f*)(scores + off));
    const v4f wv = *(const v4f*)(&sw[wkb]);   // ds_load_b128
    const float base = sh[hk] + st[tk];

    v4f o;
    o.x = s.x + base + wv.x;
    o.y = s.y + base + wv.y;
    o.z = s.z + base + wv.z;
    o.w = s.w + base + wv.w;
    __builtin_nontemporal_store(o, (v4f*)(out + off));
}

extern "C" void kernel_launch(void* const* d_in, const int* in_sizes, int n_in,
                              void* d_out, int out_size, void* d_ws, size_t ws_size,
                              hipStream_t stream) {
    const float* q      = (const float*)d_in[0];  // [2,8,2048,64]
    const float* scores = (const float*)d_in[1];  // [2,8,2048,2048]
    const float* hemb   = (const float*)d_in[2];  // [31,64]
    const float* wemb   = (const float*)d_in[3];  // [31,64]
    const float* temb   = (const float*)d_in[4];  // [15,64]
    float* out = (float*)d_out;

    float* ws   = (float*)d_ws;            // needs 5 MB: 2+2+1
    float* relh = ws;                      // [32768,16]
    float* relw = ws + (long)NROWS * 16;   // [32768,16]
    float* relt = ws + (long)NROWS * 32;   // [32768, 8]

    rel_wmma_kernel<<<512, 256, 0, stream>>>(q, hemb, wemb, temb, relh, relw, relt);
    add_bias_kernel<<<NROWS, 512, 0, stream>>>(scores, relh, relw, relt, out);
}